// MultiLabelEmbedding_6794638262887
// MI455X (gfx1250) — compile-verified
//
#include <hip/hip_runtime.h>
#include <stdint.h>

typedef unsigned int v4u __attribute__((ext_vector_type(4)));
typedef int          v4i __attribute__((ext_vector_type(4)));
typedef int          v8i __attribute__((ext_vector_type(8)));

namespace {
constexpr int kB = 16384;
constexpr int kL = 50;
constexpr int kD = 64;                 // row = 256 bytes
constexpr int kWavesPerBlock = 8;      // 256 threads = 8 wave32
constexpr int kRowsPerTdm = 8;         // 32-bit gather indices -> 8 rows / TDM op
constexpr int kBufFloats = kRowsPerTdm * kD;   // 512 floats = 2 KB per buffer
}

__global__ __launch_bounds__(kWavesPerBlock * 32)
void embag_tdm_kernel(const float* __restrict__ table,
                      const int*   __restrict__ indices,
                      const int*   __restrict__ lengths,
                      float*       __restrict__ out)
{
    // Per-wave double-buffered staging: 8 waves * 2 * 2KB = 32 KB LDS.
    __shared__ float smem[kWavesPerBlock * 2 * kBufFloats];

    const int lane = threadIdx.x & 31;
    const int wave = threadIdx.x >> 5;
    // Wave-uniform bag id, made visibly uniform to the compiler so that the
    // index/length fetches can lower to scalar (s_load) reads that feed the
    // TDM descriptor SGPR groups directly.
    const int bag = __builtin_amdgcn_readfirstlane(
        (int)(blockIdx.x * kWavesPerBlock + wave));

    int len = __builtin_amdgcn_readfirstlane(lengths[bag]);
    if (len < 0) len = 0;
    if (len > kL) len = kL;
    const int nchunks = (len + kRowsPerTdm - 1) / kRowsPerTdm;

    // LDS byte offset of this wave's staging area (low 32 bits of the
    // generic shared pointer are the LDS offset on gfx1250).
    float* wbuf = &smem[wave * 2 * kBufFloats];
    const unsigned lds0 =
        (unsigned)__builtin_amdgcn_readfirstlane((int)(unsigned)(uintptr_t)wbuf);

    // D# group0 words 2/3: 57-bit global base address of the table, type=2.
    const uint64_t tab  = (uint64_t)(uintptr_t)table;
    const unsigned gaLo = (unsigned)tab;
    const unsigned gaHi = ((unsigned)(tab >> 32) & 0x01FFFFFFu) | 0x80000000u;

    const int* bagIdx = indices + bag * kL;

    // Issue one TDM gather of up to 8 table rows into buffer (c & 1).
    auto issue_chunk = [&](int c) {
        const int base = c * kRowsPerTdm;
        int nv = len - base;
        if (nv > kRowsPerTdm) nv = kRowsPerTdm;
        int ri[8];
#pragma unroll
        for (int k = 0; k < 8; ++k) {
            const int off = base + ((k < nv) ? k : 0);   // clamp: stay in-bounds
            ri[k] = __builtin_amdgcn_readfirstlane(bagIdx[off]);
        }
        const unsigned ldsAddr = lds0 + (unsigned)((c & 1) * (kBufFloats * 4));
        // Group 0: count=1 | gather_index_size=32b (bit30) | gather_mode (bit31)
        v4u g0 = { 0xC0000001u, ldsAddr, gaLo, gaHi };
        // Group 1:
        //  w0: data_size=4B (bits17:16)
        //  w1: tensor_dim0 = 64   (bits63:48)
        //  w2: tensor_dim1 lo16 = 0x86A0 (100000) in bits95:80
        //  w3: tensor_dim1 hi16 = 1, tile_dim0 = 64 (bits127:112)
        //  w4: tile_dim1 = #valid gather indices
        //  w5: tensor_dim0_stride = 64 elements
        v8i g1 = { (int)0x00020000, (int)0x00400000, (int)0x86A00000,
                   (int)0x00400001, nv, (int)kD, 0, 0 };
        v4i g2 = { ri[0], ri[1], ri[2], ri[3] };   // gather row indices 0..3
        v4i g3 = { ri[4], ri[5], ri[6], ri[7] };   // gather row indices 4..7
        v8i g4 = { 0, 0, 0, 0, 0, 0, 0, 0 };       // unused/reserved group (VADDR4)
        __builtin_amdgcn_tensor_load_to_lds(g0, g1, g2, g3, g4, 0);
    };

    float accx = 0.f, accy = 0.f;

    if (nchunks > 0) issue_chunk(0);

    for (int c = 0; c < nchunks; ++c) {
        if (c + 1 < nchunks) {
            issue_chunk(c + 1);                       // prefetch next chunk
            __builtin_amdgcn_s_wait_tensorcnt(1);     // chunk c has landed
        } else {
            __builtin_amdgcn_s_wait_tensorcnt(0);     // drain
        }
        const float2* rows =
            (const float2*)&smem[wave * 2 * kBufFloats + (c & 1) * kBufFloats];
        int nv = len - c * kRowsPerTdm;
        if (nv > kRowsPerTdm) nv = kRowsPerTdm;
#pragma unroll
        for (int r = 0; r < kRowsPerTdm; ++r) {
            if (r < nv) {
                const float2 v = rows[r * (kD / 2) + lane];
                accx += v.x;
                accy += v.y;
            }
        }
    }

    const float inv = 1.0f / (float)(len > 0 ? len : 1);
    float2 res;
    res.x = accx * inv;
    res.y = accy * inv;
    ((float2*)(out + (size_t)bag * kD))[lane] = res;   // coalesced b64 store
}

extern "C" void kernel_launch(void* const* d_in, const int* in_sizes, int n_in,
                              void* d_out, int out_size, void* d_ws, size_t ws_size,
                              hipStream_t stream) {
    const float* table   = (const float*)d_in[0];
    const int*   indices = (const int*)d_in[1];
    const int*   lengths = (const int*)d_in[2];
    float*       out     = (float*)d_out;

    dim3 block(kWavesPerBlock * 32);        // 256 threads = 8 wave32
    dim3 grid(kB / kWavesPerBlock);         // 2048 blocks, one wave per bag
    embag_tdm_kernel<<<grid, block, 0, stream>>>(table, indices, lengths, out);
}